// Masked_Attention_69011534512864
// MI455X (gfx1250) — compile-verified
//
#include <hip/hip_runtime.h>

typedef __attribute__((ext_vector_type(2))) float v2f;
typedef __attribute__((ext_vector_type(4))) float v4f;
typedef __attribute__((ext_vector_type(8))) float v8f;
typedef __attribute__((ext_vector_type(2))) int   v2i;

typedef __attribute__((address_space(1))) v2i g_v2i;  // global
typedef __attribute__((address_space(3))) v2i l_v2i;  // LDS

static constexpr int BATCH = 2;
static constexpr int SEQ   = 2048;
static constexpr int EMB   = 1024;
static constexpr int NH    = 16;
static constexpr int HD    = 64;

__device__ __forceinline__ v8f wmma_f32(v2f a, v2f b, v8f c) {
  // D = A(16x4 f32) * B(4x16 f32) + C(16x16 f32)
  return __builtin_amdgcn_wmma_f32_16x16x4_f32(
      /*neg_a=*/false, a, /*neg_b=*/false, b,
      /*c_mod=*/(short)0, c, /*reuse_a=*/false, /*reuse_b=*/false);
}

// Async point-to-point global->LDS copy (8 bytes per lane), ASYNCcnt-tracked.
__device__ __forceinline__ void async_b64(const float* g, float* l) {
  __builtin_amdgcn_global_load_async_to_lds_b64(
      (g_v2i*)g, (l_v2i*)l, /*offset=*/0, /*cpol=*/0);
}

// ---------------------------------------------------------------------------
// GEMM: out = X[M,1024] @ W[1024,1024] + bias
// LAYOUT 0: out[m*1024 + n]            (plain row-major, final projection)
// LAYOUT 1: out[((b*NH+h)*SEQ+s)*HD+d] (QKV, b=m/SEQ, s=m%SEQ, h=n/HD, d=n%HD)
// Block: 256 threads (8 waves), 64x64 output tile, K-tile 32.
// Double-buffered async global->LDS staging: one barrier per K-tile, copy of
// tile k+1 fully overlapped with WMMA on tile k.
// ---------------------------------------------------------------------------
constexpr int BM = 64, BN = 64, BK = 32;
constexpr int XP = BK + 2;  // 34 floats: 136B rows -> b64-aligned, conflict-free
constexpr int WP = BN + 2;  // 66 floats: 264B rows -> b64-aligned

__device__ __forceinline__ void stage_tile_async(
    const float* __restrict__ X, const float* __restrict__ W,
    float* xsb, float* wsb, int rowX, int colW, int kb, int lr, int lc)
{
  const float* xsrc = X + (size_t)(rowX + lr) * EMB + kb + lc;
  const float* wsrc = W + (size_t)(kb + lr) * EMB + colW + lc;
  async_b64(xsrc,                        &xsb[lr * XP + lc]);
  async_b64(xsrc + 2,                    &xsb[lr * XP + lc + 2]);
  async_b64(xsrc + (size_t)32 * EMB,     &xsb[(lr + 32) * XP + lc]);
  async_b64(xsrc + (size_t)32 * EMB + 2, &xsb[(lr + 32) * XP + lc + 2]);
  async_b64(wsrc,                        &wsb[lr * WP + lc]);
  async_b64(wsrc + 2,                    &wsb[lr * WP + lc + 2]);
  async_b64(wsrc + 32,                   &wsb[lr * WP + lc + 32]);
  async_b64(wsrc + 34,                   &wsb[lr * WP + lc + 34]);
}

template <int LAYOUT>
__global__ __launch_bounds__(256) void gemm_bias_kernel(
    const float* __restrict__ X, const float* __restrict__ W,
    const float* __restrict__ bias, float* __restrict__ out)
{
  __shared__ float xs[2][BM * XP];
  __shared__ float ws[2][BK * WP];

  const int tid  = threadIdx.x;
  const int wave = tid >> 5;
  const int lane = tid & 31;
  const int r  = lane & 15;
  const int hi = lane >> 4;
  const int m0 = (wave & 3) * 16;
  const int n0 = (wave >> 2) * 32;
  const int rowX = blockIdx.x * BM;
  const int colW = blockIdx.y * BN;
  const int lr = tid >> 3;        // 0..31
  const int lc = (tid & 7) * 4;   // 0,4,..,28

  v8f acc0 = {};
  v8f acc1 = {};

  // prologue: async-stage first K-tile into buffer 0
  stage_tile_async(X, W, xs[0], ws[0], rowX, colW, 0, lr, lc);

  int buf = 0;
  for (int kb = 0; kb < EMB; kb += BK, buf ^= 1) {
    asm volatile("s_wait_asynccnt 0" ::: "memory");
    __syncthreads();  // all waves' async writes for buf are visible

    if (kb + BK < EMB)
      stage_tile_async(X, W, xs[buf ^ 1], ws[buf ^ 1], rowX, colW, kb + BK, lr, lc);

    const float* xsb = xs[buf];
    const float* wsb = ws[buf];
#pragma unroll
    for (int kk = 0; kk < BK; kk += 4) {
      const int ka = kk + 2 * hi;
      v2f a;
      a.x = xsb[(m0 + r) * XP + ka];
      a.y = xsb[(m0 + r) * XP + ka + 1];
      v2f b0;
      b0.x = wsb[ka * WP + n0 + r];
      b0.y = wsb[(ka + 1) * WP + n0 + r];
      v2f b1;
      b1.x = wsb[ka * WP + n0 + 16 + r];
      b1.y = wsb[(ka + 1) * WP + n0 + 16 + r];
      acc0 = wmma_f32(a, b0, acc0);
      acc1 = wmma_f32(a, b1, acc1);
    }
  }

#pragma unroll
  for (int v = 0; v < 8; v++) {
    const int row = rowX + m0 + v + 8 * hi;
    const int c0  = colW + n0 + r;
    const int c1  = c0 + 16;
    const float o0 = acc0[v] + bias[c0];
    const float o1 = acc1[v] + bias[c1];
    if (LAYOUT == 0) {
      out[(size_t)row * EMB + c0] = o0;
      out[(size_t)row * EMB + c1] = o1;
    } else {
      const int bb = row >> 11;      // / SEQ
      const int s  = row & (SEQ - 1);
      const int h0 = c0 >> 6, d0 = c0 & 63;
      const int h1 = c1 >> 6, d1 = c1 & 63;
      out[((size_t)(bb * NH + h0) * SEQ + s) * HD + d0] = o0;
      out[((size_t)(bb * NH + h1) * SEQ + s) * HD + d1] = o1;
    }
  }
}

// ---------------------------------------------------------------------------
// Flash attention (causal), fp32 WMMA.
// Grid: (SEQ/128, NH, BATCH). Block: 256 (8 waves).
// Each wave owns a 16-query stripe; Q fragments in registers; online softmax
// via shfl_xor within 16-lane halves; P transposed through a per-wave LDS
// stripe (wave32 lockstep + s_wait_dscnt -> no workgroup barrier needed).
// ---------------------------------------------------------------------------
__global__ __launch_bounds__(256) void flash_attn_kernel(
    const float* __restrict__ Q, const float* __restrict__ K,
    const float* __restrict__ V, float* __restrict__ out)
{
  constexpr int PST = 66;  // padded row stride (floats) for P stripe
  __shared__ float Pst[8 * 16 * PST];

  const int tid  = threadIdx.x;
  const int wave = tid >> 5;
  const int lane = tid & 31;
  const int r  = lane & 15;
  const int hi = lane >> 4;
  const int b = blockIdx.z, h = blockIdx.y;
  const int q0 = blockIdx.x * 128 + wave * 16;

  const size_t headoff = (size_t)(b * NH + h) * SEQ * HD;
  const float* Qh = Q + headoff;
  const float* Kh = K + headoff;
  const float* Vh = V + headoff;
  float* pw = &Pst[wave * 16 * PST];

  // Q fragments for this wave's stripe, pre-scaled by 1/sqrt(D) = 0.125
  v2f qf[16];
  {
    const float* qp = Qh + (size_t)(q0 + r) * HD + 2 * hi;
#pragma unroll
    for (int ks = 0; ks < 16; ks++) {
      qf[ks].x = qp[ks * 4]     * 0.125f;
      qf[ks].y = qp[ks * 4 + 1] * 0.125f;
    }
  }

  v8f oc[4];
#pragma unroll
  for (int dt = 0; dt < 4; dt++) oc[dt] = (v8f){};
  float mrun[8], lrun[8];
#pragma unroll
  for (int v = 0; v < 8; v++) { mrun[v] = -3.0e38f; lrun[v] = 0.0f; }

  const int nkt = (q0 + 16 + 63) >> 6;  // causal trip count for this wave
  for (int kt = 0; kt < nkt; kt++) {
    const int k0 = kt * 64;

    // S = (Q*scale) @ K^T  : 4 N-tiles of 16 keys, K-dim = 64 (d)
    v8f sc[4];
#pragma unroll
    for (int nt = 0; nt < 4; nt++) sc[nt] = (v8f){};
#pragma unroll
    for (int ks = 0; ks < 16; ks++) {
      const v2f a = qf[ks];
      const int kd = ks * 4 + 2 * hi;
      const float* kp = Kh + (size_t)(k0 + r) * HD + kd;
#pragma unroll
      for (int nt = 0; nt < 4; nt++) {
        v2f bb;
        bb.x = kp[nt * 16 * HD];
        bb.y = kp[nt * 16 * HD + 1];
        sc[nt] = wmma_f32(a, bb, sc[nt]);
      }
    }

    // causal mask: key > query -> -inf
#pragma unroll
    for (int nt = 0; nt < 4; nt++) {
      const int kcol = k0 + nt * 16 + r;
#pragma unroll
      for (int v = 0; v < 8; v++) {
        const int qrow = q0 + v + 8 * hi;
        if (kcol > qrow) sc[nt][v] = -3.0e38f;
      }
    }

    // online softmax, per row (row v+8*hi lives in 16 lanes of one half)
#pragma unroll
    for (int v = 0; v < 8; v++) {
      float mx = fmaxf(fmaxf(sc[0][v], sc[1][v]), fmaxf(sc[2][v], sc[3][v]));
#pragma unroll
      for (int off = 1; off < 16; off <<= 1)
        mx = fmaxf(mx, __shfl_xor(mx, off, 32));
      const float mnew  = fmaxf(mrun[v], mx);
      const float alpha = __expf(mrun[v] - mnew);
      mrun[v] = mnew;
      float rs = 0.0f;
#pragma unroll
      for (int nt = 0; nt < 4; nt++) {
        const float p = __expf(sc[nt][v] - mnew);
        sc[nt][v] = p;
        rs += p;
      }
#pragma unroll
      for (int off = 1; off < 16; off <<= 1)
        rs += __shfl_xor(rs, off, 32);
      lrun[v] = lrun[v] * alpha + rs;
#pragma unroll
      for (int dt = 0; dt < 4; dt++) oc[dt][v] *= alpha;
    }

    // Transpose P through this wave's private LDS stripe (C-layout -> A-layout)
#pragma unroll
    for (int nt = 0; nt < 4; nt++)
#pragma unroll
      for (int v = 0; v < 8; v++)
        pw[(v + 8 * hi) * PST + nt * 16 + r] = sc[nt][v];
    asm volatile("s_wait_dscnt 0" ::: "memory");

    // O += P @ V  : 4 D-tiles of 16, K-dim = 64 (keys)
#pragma unroll
    for (int ks = 0; ks < 16; ks++) {
      const int kk = ks * 4 + 2 * hi;
      v2f a;
      a.x = pw[r * PST + kk];
      a.y = pw[r * PST + kk + 1];
      const float* vp = Vh + (size_t)(k0 + kk) * HD;
#pragma unroll
      for (int dt = 0; dt < 4; dt++) {
        v2f bb;
        bb.x = vp[dt * 16 + r];
        bb.y = vp[HD + dt * 16 + r];
        oc[dt] = wmma_f32(a, bb, oc[dt]);
      }
    }
  }

  // normalize and write attn output in [B, S, E] layout (E = h*HD + d)
#pragma unroll
  for (int v = 0; v < 8; v++) {
    const float inv = 1.0f / lrun[v];
    const int s = q0 + v + 8 * hi;
    float* op = out + ((size_t)b * SEQ + s) * EMB + h * HD;
#pragma unroll
    for (int dt = 0; dt < 4; dt++)
      op[dt * 16 + r] = oc[dt][v] * inv;
  }
}

// ---------------------------------------------------------------------------
extern "C" void kernel_launch(void* const* d_in, const int* in_sizes, int n_in,
                              void* d_out, int out_size, void* d_ws, size_t ws_size,
                              hipStream_t stream) {
  (void)in_sizes; (void)n_in; (void)out_size; (void)ws_size;
  const float* x  = (const float*)d_in[0];
  const float* Wq = (const float*)d_in[1];
  const float* bq = (const float*)d_in[2];
  const float* Wk = (const float*)d_in[3];
  const float* bk = (const float*)d_in[4];
  const float* Wv = (const float*)d_in[5];
  const float* bv = (const float*)d_in[6];
  const float* Wo = (const float*)d_in[7];
  const float* bo = (const float*)d_in[8];
  float* out = (float*)d_out;

  const size_t per = (size_t)BATCH * NH * SEQ * HD;  // 4M floats = 16 MB
  float* Qw = (float*)d_ws;
  float* Kw = Qw + per;
  float* Vw = Kw + per;
  float* Aw = Vw + per;  // attention output, [B,S,E]

  dim3 gridG((BATCH * SEQ) / 64, EMB / 64);  // (64, 16)
  gemm_bias_kernel<1><<<gridG, 256, 0, stream>>>(x, Wq, bq, Qw);
  gemm_bias_kernel<1><<<gridG, 256, 0, stream>>>(x, Wk, bk, Kw);
  gemm_bias_kernel<1><<<gridG, 256, 0, stream>>>(x, Wv, bv, Vw);

  dim3 gridA(SEQ / 128, NH, BATCH);
  flash_attn_kernel<<<gridA, 256, 0, stream>>>(Qw, Kw, Vw, Aw);

  gemm_bias_kernel<0><<<gridG, 256, 0, stream>>>(Aw, Wo, bo, out);
}